// GravNet_2138893713600
// MI455X (gfx1250) — compile-verified
//
#include <hip/hip_runtime.h>
#include <stdint.h>

// Problem constants (from reference)
#define B_    16
#define V_    4096
#define F_    32
#define KNBR  40
#define NDIM  4
#define NFILT 48
#define NPROP 22
#define KOUT  (F_ + 2*NPROP)   // 76
#define NV    (B_*V_)          // 65536

typedef float v2f __attribute__((ext_vector_type(2)));
typedef float v8f __attribute__((ext_vector_type(8)));

// ---------------------------------------------------------------------------
// Kernel A: features = x@W_flr + b_flr (NV x 22), coords = x@W_s + b_s (NV x 4)
// Also copies x into updated[:, 0:32]. One thread per vertex; weights in LDS.
// ---------------------------------------------------------------------------
__global__ __launch_bounds__(256) void proj_kernel(
    const float* __restrict__ x,
    const float* __restrict__ Wf, const float* __restrict__ bf,
    const float* __restrict__ Ws, const float* __restrict__ bs,
    float* __restrict__ feats, float4* __restrict__ coords,
    float* __restrict__ updated)
{
  __shared__ float sWf[F_*NPROP];
  __shared__ float sWs[F_*NDIM];
  __shared__ float sbf[NPROP];
  __shared__ float sbs[NDIM];
  const int t = threadIdx.x;
  for (int i = t; i < F_*NPROP; i += 256) sWf[i] = Wf[i];
  for (int i = t; i < F_*NDIM;  i += 256) sWs[i] = Ws[i];
  if (t < NPROP) sbf[t] = bf[t];
  if (t < NDIM)  sbs[t] = bs[t];
  __syncthreads();

  const int row = blockIdx.x*256 + t;              // 0..NV-1
  const float* xr = x + (size_t)row*F_;
  float xv[F_];
  #pragma unroll
  for (int k = 0; k < F_; k += 4) {
    float4 v4 = *(const float4*)(xr + k);
    xv[k] = v4.x; xv[k+1] = v4.y; xv[k+2] = v4.z; xv[k+3] = v4.w;
  }
  // updated[:, 0:32] = x
  float* ur = updated + (size_t)row*KOUT;
  #pragma unroll
  for (int k = 0; k < F_; k += 4)
    *(float4*)(ur + k) = make_float4(xv[k], xv[k+1], xv[k+2], xv[k+3]);

  float* fr = feats + (size_t)row*NPROP;
  #pragma unroll
  for (int n = 0; n < NPROP; n++) {
    float acc = sbf[n];
    #pragma unroll
    for (int k = 0; k < F_; k++) acc = fmaf(xv[k], sWf[k*NPROP + n], acc);
    fr[n] = acc;
  }
  float c[NDIM];
  #pragma unroll
  for (int n = 0; n < NDIM; n++) {
    float acc = sbs[n];
    #pragma unroll
    for (int k = 0; k < F_; k++) acc = fmaf(xv[k], sWs[k*NDIM + n], acc);
    c[n] = acc;
  }
  coords[row] = make_float4(c[0], c[1], c[2], c[3]);
}

// ---------------------------------------------------------------------------
// Kernel B: fused kNN (top-40 smallest squared distances, self at slot 0) +
// weighted max/mean aggregation over 39 neighbors -> updated[:, 32:76].
// Coords of one batch (64 KB) staged to LDS with GLOBAL_LOAD_ASYNC_TO_LDS_B128
// (CDNA5 async engine, ASYNCcnt) -- no VGPR round-trip. Per-thread top-k lists
// in LDS, lane-interleaved (entry i of thread t at [i*256 + t]) so same-slot
// accesses are conflict-free. dmat (1.07 GB) is never materialized.
// Dynamic LDS: 4096*16 + 40*256*4 + 40*256*4 = 147456 bytes.
// ---------------------------------------------------------------------------
__global__ __launch_bounds__(256) void knn_agg_kernel(
    const float4* __restrict__ coords, const float* __restrict__ feats,
    float* __restrict__ updated)
{
  extern __shared__ float smem[];
  float4* sc   = (float4*)smem;                    // V_ coords
  float*  topd = smem + 4*V_;                      // KNBR*256
  int*    topi = (int*)(smem + 4*V_ + KNBR*256);   // KNBR*256

  const int t      = threadIdx.x;
  const int b      = blockIdx.x / (V_/256);
  const int vchunk = blockIdx.x % (V_/256);

  // ---- async stage: coords[b, :] -> LDS (16 x B128 per thread) ----
  {
    const float4* cb = coords + (size_t)b*V_;
    #pragma unroll
    for (int i = 0; i < V_/256; i++) {             // 16 iterations
      const int idx = t + i*256;                   // float4 index
      // flat addr[31:0] of an LDS location == wave-relative LDS byte address
      uint32_t lds_off = (uint32_t)(uintptr_t)(&sc[idx]);
      const void* gaddr = (const void*)(cb + idx);
      asm volatile("global_load_async_to_lds_b128 %0, %1, off"
                   :: "v"(lds_off), "v"(gaddr)
                   : "memory");
    }
    asm volatile("s_wait_asynccnt 0x0" ::: "memory");
  }
  __syncthreads();

  const int v = vchunk*256 + t;
  const float4 cv = sc[v];

  #pragma unroll
  for (int i = 0; i < KNBR; i++) { topd[i*256 + t] = 3.0e38f; topi[i*256 + t] = -1; }
  float worst = 3.0e38f;

  for (int w = 0; w < V_; w++) {
    float4 cw = sc[w];                              // LDS broadcast
    float dx = cv.x - cw.x, dy = cv.y - cw.y;
    float dz = cv.z - cw.z, dq = cv.w - cw.w;
    float d = dx*dx + dy*dy + dz*dz + dq*dq;
    if (d < worst) {                                // rare: sorted insert
      int i = KNBR - 1;
      while (i > 0 && topd[(i-1)*256 + t] > d) {    // strict '>' keeps lower idx on ties
        topd[i*256 + t] = topd[(i-1)*256 + t];
        topi[i*256 + t] = topi[(i-1)*256 + t];
        i--;
      }
      topd[i*256 + t] = d; topi[i*256 + t] = w;
      worst = topd[(KNBR-1)*256 + t];
    }
  }

  // Aggregate neighbors 1..39 (slot 0 == self, dropped like idx[:, :, 1:]).
  // Feature rows are 88 B (8-byte aligned) -> gather as 11 x float2.
  float mx[NPROP], sm[NPROP];
  #pragma unroll
  for (int n = 0; n < NPROP; n++) { mx[n] = -3.0e38f; sm[n] = 0.0f; }
  for (int j = 1; j < KNBR; j++) {
    float dj  = topd[j*256 + t];
    int   ij  = topi[j*256 + t];
    float wgt = __expf(-10.0f * fabsf(dj));
    const float2* fj2 = (const float2*)(feats + ((size_t)b*V_ + ij)*NPROP);
    #pragma unroll
    for (int n2 = 0; n2 < NPROP/2; n2++) {
      float2 f2 = fj2[n2];
      float f0 = wgt * f2.x;
      float f1 = wgt * f2.y;
      mx[2*n2]   = fmaxf(mx[2*n2],   f0);
      mx[2*n2+1] = fmaxf(mx[2*n2+1], f1);
      sm[2*n2]   += f0;
      sm[2*n2+1] += f1;
    }
  }
  float* ur = updated + ((size_t)b*V_ + v)*KOUT;
  const float inv = 1.0f / (float)(KNBR - 1);
  #pragma unroll
  for (int n = 0; n < NPROP; n++) {
    ur[F_ + n]         = mx[n];
    ur[F_ + NPROP + n] = sm[n] * inv;
  }
}

// ---------------------------------------------------------------------------
// Kernel C: out = tanh(updated @ W_out + b_out) via V_WMMA_F32_16X16X4_F32.
// K=76 = 19 chained k=4 steps (exact), N=48 = 3 tiles of 16. One wave per
// 16-row M-tile, A fragment reused across the 3 N accumulators. Full waves
// only (EXEC all ones, as WMMA requires).
// A 16x4 layout: lanes 0-15 hold A[m][0],A[m][1]; lanes 16-31 A[m][2],A[m][3].
// B 4x16 layout: v0/v1 hold rows {2*kh, 2*kh+1} at column n = lane&15.
// C/D layout: VGPR i <-> rows i (lanes 0-15) and i+8 (lanes 16-31).
// ---------------------------------------------------------------------------
__global__ __launch_bounds__(256) void out_gemm_kernel(
    const float* __restrict__ updated, const float* __restrict__ Wout,
    const float* __restrict__ bout, float* __restrict__ out)
{
  __shared__ float sW[KOUT*NFILT];   // 76*48*4 = 14592 B
  __shared__ float sb[NFILT];
  const int t = threadIdx.x;
  for (int i = t; i < KOUT*NFILT; i += 256) sW[i] = Wout[i];
  if (t < NFILT) sb[t] = bout[t];
  __syncthreads();

  const int lane = t & 31;
  const int wave = t >> 5;
  const int m    = lane & 15;
  const int kh   = lane >> 4;                      // 0 or 1 (K half)
  const int rowBase = blockIdx.x*128 + wave*16;

  const float* ar = updated + (size_t)(rowBase + m)*KOUT + 2*kh;
  v8f c0 = {}, c1 = {}, c2 = {};
  #pragma unroll
  for (int kk = 0; kk < 19; kk++) {
    v2f a;
    a.x = ar[4*kk + 0];
    a.y = ar[4*kk + 1];
    const int k0 = 4*kk + 2*kh;
    v2f b0, b1, b2;
    b0.x = sW[k0*NFILT + m];        b0.y = sW[(k0+1)*NFILT + m];
    b1.x = sW[k0*NFILT + 16 + m];   b1.y = sW[(k0+1)*NFILT + 16 + m];
    b2.x = sW[k0*NFILT + 32 + m];   b2.y = sW[(k0+1)*NFILT + 32 + m];
    c0 = __builtin_amdgcn_wmma_f32_16x16x4_f32(false, a, false, b0, (short)0, c0, false, false);
    c1 = __builtin_amdgcn_wmma_f32_16x16x4_f32(false, a, false, b1, (short)0, c1, false, false);
    c2 = __builtin_amdgcn_wmma_f32_16x16x4_f32(false, a, false, b2, (short)0, c2, false, false);
  }

  #pragma unroll
  for (int i = 0; i < 8; i++) {
    const int row = rowBase + i + 8*kh;
    float vals[3] = { c0[i], c1[i], c2[i] };
    #pragma unroll
    for (int nt = 0; nt < 3; nt++) {
      const int col = nt*16 + m;
      out[(size_t)row*NFILT + col] = tanhf(vals[nt] + sb[col]);
    }
  }
}

// ---------------------------------------------------------------------------
// Workspace layout (26.7 MB total):
//   feats   : NV*NPROP floats  (5,767,168 B)
//   coords  : NV float4        (1,048,576 B)
//   updated : NV*KOUT floats   (19,922,944 B)
// ---------------------------------------------------------------------------
extern "C" void kernel_launch(void* const* d_in, const int* in_sizes, int n_in,
                              void* d_out, int out_size, void* d_ws, size_t ws_size,
                              hipStream_t stream) {
  (void)in_sizes; (void)n_in; (void)out_size; (void)ws_size;
  const float* x  = (const float*)d_in[0];
  const float* Wf = (const float*)d_in[1];
  const float* bf = (const float*)d_in[2];
  const float* Ws = (const float*)d_in[3];
  const float* bs = (const float*)d_in[4];
  const float* Wo = (const float*)d_in[5];
  const float* bo = (const float*)d_in[6];
  float* out = (float*)d_out;

  char* ws = (char*)d_ws;
  float*  feats   = (float*)ws;
  float4* coords  = (float4*)(ws + sizeof(float)*(size_t)NV*NPROP);
  float*  updated = (float*)(ws + sizeof(float)*(size_t)NV*NPROP + sizeof(float4)*(size_t)NV);

  proj_kernel<<<NV/256, 256, 0, stream>>>(x, Wf, bf, Ws, bs, feats, coords, updated);

  const size_t smem = (size_t)(4*V_ + 2*KNBR*256) * sizeof(float);  // 147456 B
  knn_agg_kernel<<<B_*(V_/256), 256, smem, stream>>>(coords, feats, updated);

  out_gemm_kernel<<<NV/128, 256, 0, stream>>>(updated, Wo, bo, out);
}